// AdaptivePrototypeRefiner_28432683500081
// MI455X (gfx1250) — compile-verified
//
#include <hip/hip_runtime.h>
#include <math.h>

// Problem constants (match reference)
#define KP 32        // prototypes
#define NQ 16384     // queries
#define CD 512       // channels
#define HREF 512
#define HCONF 256
#define NSTEP 3

typedef __attribute__((ext_vector_type(2))) float v2f;
typedef __attribute__((ext_vector_type(8))) float v8f;

// D = A(16x4) * B(4x16) + C, full fp32 via CDNA5 WMMA
__device__ __forceinline__ v8f wmma4(v2f a, v2f b, v8f c) {
  return __builtin_amdgcn_wmma_f32_16x16x4_f32(
      /*neg_a=*/false, a, /*neg_b=*/false, b,
      /*c_mod=*/(short)0, c, /*reuse_a=*/false, /*reuse_b=*/false);
}

__device__ __forceinline__ float wave_max(float x) {
  for (int off = 16; off > 0; off >>= 1) x = fmaxf(x, __shfl_xor(x, off, 32));
  return x;
}
__device__ __forceinline__ float wave_sum(float x) {
  for (int off = 16; off > 0; off >>= 1) x += __shfl_xor(x, off, 32);
  return x;
}

// ---------------- init: zero accumulators, seed refined = prototypes ----------
__global__ void k_init(const float* __restrict__ protos, float* __restrict__ out,
                       float* __restrict__ S, float* __restrict__ wsum,
                       float* __restrict__ confsum) {
  int i = blockIdx.x * blockDim.x + threadIdx.x;
  if (i < NSTEP * KP * CD) S[i] = 0.f;
  if (i < NSTEP * KP) wsum[i] = 0.f;
  if (i < KP) confsum[i] = 0.f;
  if (i < KP * CD) out[i] = protos[i];
}

// ---------------- softmax stats + wsum ---------------------------------------
// One wave per query; lane k holds distance d[q][k] (K == 32 == wave size).
// Stores per (t,q): m = max_k(-d/T), r = 1/sum_k exp(-d/T - m).
__global__ void k_softmax(const float* __restrict__ qd, float* __restrict__ stats,
                          float* __restrict__ wsum_g) {
  __shared__ float swsum[NSTEP * KP];
  int tid = threadIdx.x;
  if (tid < NSTEP * KP) swsum[tid] = 0.f;
  __syncthreads();
  int lane = tid & 31;
  int q = blockIdx.x * (blockDim.x >> 5) + (tid >> 5);
  float d = qd[(size_t)q * KP + lane];
  for (int t = 0; t < NSTEP; ++t) {
    float x = -d / (float)(t + 1);
    float m = wave_max(x);
    float e = __expf(x - m);
    float s = wave_sum(e);
    float r = 1.f / s;
    if (lane == 0) {
      stats[((size_t)t * NQ + q) * 2 + 0] = m;
      stats[((size_t)t * NQ + q) * 2 + 1] = r;
    }
    atomicAdd(&swsum[t * KP + lane], e * r);
  }
  __syncthreads();
  if (tid < NSTEP * KP) atomicAdd(&wsum_g[tid], swsum[tid]);
}

// ---------------- S[t] = soft[t].T @ QF  (32 x 512 per step) ------------------
// wave id -> (t, rowTile(2), colTile(32), kChunk(16)); soft recomputed from qd.
__global__ void k_wmean_gemm(const float* __restrict__ qd,
                             const float* __restrict__ stats,
                             const float* __restrict__ qf,
                             float* __restrict__ S) {
  int wid = (blockIdx.x * blockDim.x + threadIdx.x) >> 5;
  int lane = threadIdx.x & 31;
  int n = lane & 15, half = lane >> 4;
  int kc = wid & 15;
  int ct = (wid >> 4) & 31;
  int rt = (wid >> 9) & 1;
  int t  = wid >> 10;
  float invT = 1.f / (float)(t + 1);
  const float* st = stats + (size_t)t * NQ * 2;
  int p = rt * 16 + (lane & 15);
  int c = ct * 16 + n;
  v8f acc = {};
  int q0 = kc * (NQ / 16), q1 = q0 + (NQ / 16);
  for (int q = q0; q < q1; q += 4) {
    int ka = q + half * 2;
    float m0 = st[(size_t)ka * 2], r0 = st[(size_t)ka * 2 + 1];
    float m1 = st[(size_t)(ka + 1) * 2], r1 = st[(size_t)(ka + 1) * 2 + 1];
    v2f a, b;
    a.x = __expf(-qd[(size_t)ka * KP + p] * invT - m0) * r0;
    a.y = __expf(-qd[(size_t)(ka + 1) * KP + p] * invT - m1) * r1;
    b.x = qf[(size_t)ka * CD + c];
    b.y = qf[(size_t)(ka + 1) * CD + c];
    acc = wmma4(a, b, acc);
  }
  float* Sp = S + (size_t)t * KP * CD;
  #pragma unroll
  for (int v = 0; v < 8; ++v) {
    int m = rt * 16 + v + half * 8;
    atomicAdd(&Sp[m * CD + c], acc[v]);
  }
}

// ---------------- hp = prototypes @ Wc1[:512]  (32 x 256) ---------------------
__global__ void k_hp(const float* __restrict__ protos, const float* __restrict__ Wc1,
                     float* __restrict__ hp) {
  int wid = (blockIdx.x * blockDim.x + threadIdx.x) >> 5;
  int lane = threadIdx.x & 31;
  int n = lane & 15, half = lane >> 4;
  int ct = wid & 15;      // 256/16 col tiles
  int rt = wid >> 4;      // 2 row tiles
  int p = rt * 16 + (lane & 15);
  int c = ct * 16 + n;
  v8f acc = {};
  for (int k0 = 0; k0 < CD; k0 += 4) {
    int ka = k0 + half * 2;
    v2f a = *(const v2f*)(protos + (size_t)p * CD + ka);
    v2f b;
    b.x = Wc1[(size_t)ka * HCONF + c];
    b.y = Wc1[(size_t)(ka + 1) * HCONF + c];
    acc = wmma4(a, b, acc);
  }
  #pragma unroll
  for (int v = 0; v < 8; ++v)
    hp[(rt * 16 + v + half * 8) * HCONF + c] = acc[v];
}

// ---------------- hq = QF @ Wc1[512:]  (16384 x 256, the big GEMM) -----------
// Each wave owns a 16x64 output slab (4 WMMA col-tiles) to reuse the A tile.
__global__ void k_hq(const float* __restrict__ qf, const float* __restrict__ Wc1,
                     float* __restrict__ hq) {
  int wid = (blockIdx.x * blockDim.x + threadIdx.x) >> 5;
  int lane = threadIdx.x & 31;
  int n = lane & 15, half = lane >> 4;
  int cg = wid & 3;       // col group of 64
  int rt = wid >> 2;      // 1024 row tiles
  int r = rt * 16 + (lane & 15);
  int cbase = cg * 64;
  const float* Wq = Wc1 + (size_t)CD * HCONF;
  v8f acc0 = {}, acc1 = {}, acc2 = {}, acc3 = {};
  for (int k0 = 0; k0 < CD; k0 += 4) {
    int ka = k0 + half * 2;
    v2f a = *(const v2f*)(qf + (size_t)r * CD + ka);
    const float* w0 = Wq + (size_t)ka * HCONF + cbase + n;
    const float* w1 = Wq + (size_t)(ka + 1) * HCONF + cbase + n;
    v2f b;
    b.x = w0[0];  b.y = w1[0];  acc0 = wmma4(a, b, acc0);
    b.x = w0[16]; b.y = w1[16]; acc1 = wmma4(a, b, acc1);
    b.x = w0[32]; b.y = w1[32]; acc2 = wmma4(a, b, acc2);
    b.x = w0[48]; b.y = w1[48]; acc3 = wmma4(a, b, acc3);
  }
  #pragma unroll
  for (int v = 0; v < 8; ++v) {
    int mm = rt * 16 + v + half * 8;
    float* o = hq + (size_t)mm * HCONF + cbase + n;
    o[0] = acc0[v]; o[16] = acc1[v]; o[32] = acc2[v]; o[48] = acc3[v];
  }
}

// ---------------- h = relu([refined, wmean_t] @ W1 + b1)  (32 x 512) ----------
__global__ void k_mlp1(const float* __restrict__ refined, const float* __restrict__ S,
                       const float* __restrict__ wsum, const float* __restrict__ W1,
                       const float* __restrict__ b1, float* __restrict__ h, int t) {
  int wid = (blockIdx.x * blockDim.x + threadIdx.x) >> 5;
  int lane = threadIdx.x & 31;
  int n = lane & 15, half = lane >> 4;
  int ct = wid & 31;
  int rt = wid >> 5;
  int p = rt * 16 + (lane & 15);
  int c = ct * 16 + n;
  const float* Sp = S + (size_t)t * KP * CD + (size_t)p * CD;
  float inv = 1.f / fmaxf(wsum[t * KP + p], 1e-6f);
  v8f acc = {};
  for (int k0 = 0; k0 < 2 * CD; k0 += 4) {
    int ka = k0 + half * 2;
    v2f a, b;
    if (ka < CD) {                     // wave-uniform branch (4 | 512)
      a.x = refined[(size_t)p * CD + ka];
      a.y = refined[(size_t)p * CD + ka + 1];
    } else {
      a.x = Sp[ka - CD] * inv;
      a.y = Sp[ka - CD + 1] * inv;
    }
    b.x = W1[(size_t)ka * HREF + c];
    b.y = W1[(size_t)(ka + 1) * HREF + c];
    acc = wmma4(a, b, acc);
  }
  float bias = b1[c];
  #pragma unroll
  for (int v = 0; v < 8; ++v)
    h[(rt * 16 + v + half * 8) * HREF + c] = fmaxf(acc[v] + bias, 0.f);
}

// ---------------- refined += 0.1 * (h @ W2 + b2)  (in place in d_out) ---------
__global__ void k_mlp2(const float* __restrict__ h, const float* __restrict__ W2,
                       const float* __restrict__ b2, float* __restrict__ refined) {
  int wid = (blockIdx.x * blockDim.x + threadIdx.x) >> 5;
  int lane = threadIdx.x & 31;
  int n = lane & 15, half = lane >> 4;
  int ct = wid & 31;
  int rt = wid >> 5;
  int p = rt * 16 + (lane & 15);
  int c = ct * 16 + n;
  v8f acc = {};
  for (int k0 = 0; k0 < HREF; k0 += 4) {
    int ka = k0 + half * 2;
    v2f a = *(const v2f*)(h + (size_t)p * HREF + ka);
    v2f b;
    b.x = W2[(size_t)ka * CD + c];
    b.y = W2[(size_t)(ka + 1) * CD + c];
    acc = wmma4(a, b, acc);
  }
  float bias = b2[c];
  #pragma unroll
  for (int v = 0; v < 8; ++v) {
    int m = rt * 16 + v + half * 8;
    refined[(size_t)m * CD + c] += 0.1f * (acc[v] + bias);
  }
}

// ---------------- confidence: mean_q sigmoid(relu(hp+hq+bc1) . Wc2 + bc2) -----
__global__ void k_conf(const float* __restrict__ hp, const float* __restrict__ hq,
                       const float* __restrict__ bc1, const float* __restrict__ Wc2,
                       const float* __restrict__ bc2, float* __restrict__ confsum) {
  __shared__ float s_hp[KP * HCONF];   // 32 KB
  __shared__ float s_conf[KP];
  int tid = threadIdx.x;
  for (int i = tid; i < KP * HCONF; i += blockDim.x) s_hp[i] = hp[i];
  if (tid < KP) s_conf[tid] = 0.f;
  __syncthreads();
  int lane = tid & 31;
  int q = blockIdx.x * (blockDim.x >> 5) + (tid >> 5);
  float hqv[8], bcv[8], wcv[8];
  #pragma unroll
  for (int j = 0; j < 8; ++j) {
    int c = lane + 32 * j;
    hqv[j] = hq[(size_t)q * HCONF + c];
    bcv[j] = bc1[c];
    wcv[j] = Wc2[c];
  }
  float b2s = bc2[0];
  for (int k = 0; k < KP; ++k) {
    float part = 0.f;
    #pragma unroll
    for (int j = 0; j < 8; ++j) {
      int c = lane + 32 * j;
      part += fmaxf(s_hp[k * HCONF + c] + hqv[j] + bcv[j], 0.f) * wcv[j];
    }
    part = wave_sum(part);
    if (lane == 0) {
      float cf = 1.f / (1.f + __expf(-(part + b2s)));
      atomicAdd(&s_conf[k], cf);
    }
  }
  __syncthreads();
  if (tid < KP) atomicAdd(&confsum[tid], s_conf[tid]);
}

__global__ void k_final(const float* __restrict__ confsum, float* __restrict__ out) {
  int k = threadIdx.x;
  if (k < KP) out[KP * CD + k] = confsum[k] * (1.f / (float)NQ);
}

extern "C" void kernel_launch(void* const* d_in, const int* in_sizes, int n_in,
                              void* d_out, int out_size, void* d_ws, size_t ws_size,
                              hipStream_t stream) {
  const float* protos = (const float*)d_in[0];
  const float* qf     = (const float*)d_in[1];
  const float* qd     = (const float*)d_in[2];
  const float* W1     = (const float*)d_in[3];
  const float* b1     = (const float*)d_in[4];
  const float* W2     = (const float*)d_in[5];
  const float* b2     = (const float*)d_in[6];
  const float* Wc1    = (const float*)d_in[7];
  const float* bc1    = (const float*)d_in[8];
  const float* Wc2    = (const float*)d_in[9];
  const float* bc2    = (const float*)d_in[10];
  float* out = (float*)d_out;

  // Workspace layout (floats), ~17.5 MB total
  float* ws      = (float*)d_ws;
  float* stats   = ws;                               // 3*NQ*2       = 98304
  float* wsum    = stats + (size_t)NSTEP * NQ * 2;   // 3*32         = 96
  float* S       = wsum + NSTEP * KP;                // 3*32*512     = 49152
  float* h       = S + (size_t)NSTEP * KP * CD;      // 32*512       = 16384
  float* hp      = h + KP * HREF;                    // 32*256       = 8192
  float* hq      = hp + KP * HCONF;                  // NQ*256       = 4194304
  float* confsum = hq + (size_t)NQ * HCONF;          // 32

  k_init<<<192, 256, 0, stream>>>(protos, out, S, wsum, confsum);
  k_softmax<<<NQ / 8, 256, 0, stream>>>(qd, stats, wsum);
  k_wmean_gemm<<<768, 128, 0, stream>>>(qd, stats, qf, S);   // 3072 waves
  k_hp<<<8, 128, 0, stream>>>(protos, Wc1, hp);              // 32 waves
  k_hq<<<1024, 128, 0, stream>>>(qf, Wc1, hq);               // 4096 waves
  for (int t = 0; t < NSTEP; ++t) {
    k_mlp1<<<16, 128, 0, stream>>>(out, S, wsum, W1, b1, h, t);
    k_mlp2<<<16, 128, 0, stream>>>(h, W2, b2, out);
  }
  k_conf<<<NQ / 8, 256, 0, stream>>>(hp, hq, bc1, Wc2, bc2, confsum);
  k_final<<<1, 32, 0, stream>>>(confsum, out);
}